// MSR_40931038331424
// MI455X (gfx1250) — compile-verified
//
#include <hip/hip_runtime.h>
#include <hip/hip_bf16.h>
#include <math.h>

// ---------------------------------------------------------------------------
// Multi-Scale Retinex, [B=8,C=3,H=1024,W=1024] f32, sigmas {15,40,80}.
// Separable Gaussian blur as banded-matrix GEMM on the CDNA5 matrix pipe
// (V_WMMA_F32_16X16X4_F32, full f32). Workload is HBM-bound (~1.3 GB moved
// at 23.3 TB/s); the f32 matrix path removes all conv FMAs from the VALU.
// Band weights live zero-padded in LDS -> branch-free operand builds; the
// chunk loop is software-pipelined so DS latency overlaps the WMMA pipe.
// ---------------------------------------------------------------------------

typedef __attribute__((ext_vector_type(2))) float v2f;
typedef __attribute__((ext_vector_type(8))) float v8f;

#define IMG_H 1024
#define IMG_W 1024
#define N_PLANES 24                     // B*C = 8*3
#define N_ELEMS (25165824LL)            // 24 * 1024 * 1024

// jnp.pad 'reflect' (no edge repeat); single reflection suffices (pad<W)
__device__ __forceinline__ int reflect_idx(int i, int n) {
    if (i < 0) i = -i;
    if (i >= n) i = 2 * n - 2 - i;
    if (i < 0) i = 0;
    if (i >= n) i = n - 1;
    return i;
}

// ---------------------------------------------------------------------------
// Gaussian coefficient init: g[i] = exp(-(i-pad)^2 / (2 sigma^2)) / sum
// scales: K=31 @ off 0, K=81 @ off 31, K=161 @ off 112 (273 floats total)
// ---------------------------------------------------------------------------
__global__ void msr_gauss_init(float* __restrict__ g) {
    __shared__ float red[256];
    const int ks[3]  = {31, 81, 161};
    const int off[3] = {0, 31, 112};
    const int tid = threadIdx.x;
    for (int s = 0; s < 3; ++s) {
        const int K = ks[s];
        const float pad = (float)(K - 1) * 0.5f;     // == sigma
        const float inv2s2 = 1.0f / (2.0f * pad * pad);
        float psum = 0.0f;
        for (int i = tid; i < K; i += 256) {
            float x = (float)i - pad;
            float e = expf(-(x * x) * inv2s2);
            g[off[s] + i] = e;
            psum += e;
        }
        red[tid] = psum;
        __syncthreads();
        for (int st = 128; st > 0; st >>= 1) {
            if (tid < st) red[tid] += red[tid + st];
            __syncthreads();
        }
        const float inv = 1.0f / red[0];
        __syncthreads();
        for (int i = tid; i < K; i += 256) g[off[s] + i] *= inv;
        __syncthreads();
    }
}

// ---------------------------------------------------------------------------
// Horizontal blur of (img + 1e-6) -> tmp.
// Workgroup: 256 threads (8 waves), tile = 16 rows x 128 cols; each wave owns
// a 16x16 output tile. Out = In_ext(16 x P) * Kband(P x 16), P chunked by 4.
// WMMA f32 16x16x4 operand layout (wave32):
//   A (16x4): lanes 0-15 = M rows; VGPR0 = {K0 | K2}, VGPR1 = {K1 | K3}
//   B (4x16): lanes = N cols;      VGPR0 = {K0 | K2}, VGPR1 = {K1 | K3}
//   C/D (16x16): VGPR r = rows {r | r+8} across half-waves
// Band operand: Kband[p, n] = g[p-n] -> read from zero-padded gpad[p-n+16],
// unconditional (no divergence).
// ---------------------------------------------------------------------------
template <int K>
__global__ __launch_bounds__(256) void msr_hblur(const float* __restrict__ img,
                                                 float* __restrict__ tmp,
                                                 const float* __restrict__ gAll,
                                                 int gOff) {
    constexpr int PAD   = K / 2;
    constexpr int NCH   = (K + 18) / 4;       // ceil((K+15)/4) chunks of 4
    constexpr int EXTA  = 112 + 4 * NCH;      // staged ext width (covers all waves)
    constexpr int HSTR  = EXTA + 1;           // odd stride -> distinct LDS banks/row
    constexpr int GPADN = 4 * NCH + 18;       // zero-padded coeff span

    __shared__ float lds[16 * HSTR];
    __shared__ float gpad[GPADN];

    const int tid  = threadIdx.x;
    const int lane = tid & 31;
    const int wave = tid >> 5;                // 0..7
    const int half = (lane >> 4) & 1;
    const int l    = lane & 15;

    const int plane = blockIdx.z;
    const int y0    = blockIdx.y * 16;
    const int xbase = blockIdx.x * 128;
    const float* src = img + (size_t)plane * IMG_H * IMG_W;
    float*       dst = tmp + (size_t)plane * IMG_H * IMG_W;

    // zero-padded coefficients: gpad[16 .. 16+K-1] = g, else 0
    for (int i = tid; i < GPADN; i += 256) {
        const int j = i - 16;
        gpad[i] = (j >= 0 && j < K) ? gAll[gOff + j] : 0.0f;
    }
    // stage 16 rows x EXTA cols with horizontal reflect padding, + 1e-6
    for (int i = tid; i < 16 * EXTA; i += 256) {
        const int r  = i / EXTA;
        const int cc = i - r * EXTA;
        const int gx = reflect_idx(xbase - PAD + cc, IMG_W);
        lds[r * HSTR + cc] = src[(size_t)(y0 + r) * IMG_W + gx] + 1e-6f;
    }
    __syncthreads();

    const int woff  = wave * 16;
    const int abase = l * HSTR + woff + 2 * half;   // + 4*ch
    const int wbase = 2 * half - l + 16;            // + 4*ch (into gpad)

    auto loadA = [&](int ch) {
        v2f a;
        a.x = lds[abase + 4 * ch];
        a.y = lds[abase + 4 * ch + 1];
        return a;
    };
    auto loadB = [&](int ch) {
        v2f b;
        b.x = gpad[wbase + 4 * ch];
        b.y = gpad[wbase + 4 * ch + 1];
        return b;
    };

    v8f acc = {};
    v2f a = loadA(0), b = loadB(0);
#pragma unroll 4
    for (int ch = 0; ch < NCH - 1; ++ch) {
        v2f an = loadA(ch + 1);
        v2f bn = loadB(ch + 1);
        acc = __builtin_amdgcn_wmma_f32_16x16x4_f32(false, a, false, b,
                                                    (short)0, acc, false, false);
        a = an; b = bn;
    }
    acc = __builtin_amdgcn_wmma_f32_16x16x4_f32(false, a, false, b,
                                                (short)0, acc, false, false);

    const int xo = xbase + woff + l;
#pragma unroll
    for (int r = 0; r < 8; ++r)
        dst[(size_t)(y0 + r + 8 * half) * IMG_W + xo] = acc[r];
}

// ---------------------------------------------------------------------------
// Vertical blur of tmp -> illumination, fused with Retinex accumulation:
//   d_acc[i] (+)= log(img[i]+1e-6) - log(ill[i]+1e-6)
// Workgroup: 256 threads (8 waves), tile = 32 out rows x 64 cols
// (wave (wy,wx): wy=wave/4 row-subtile, wx=wave%4 col-subtile).
// Here A = Kband^T (weights, from gpad), B = staged column data.
// LDS row stride 72: (72*2)%64 == 16 -> the two half-wave row reads hit
// disjoint 16-bank groups (conflict-free B fetch).
// ---------------------------------------------------------------------------
template <int K>
__global__ __launch_bounds__(256) void msr_vblur(const float* __restrict__ tmp,
                                                 const float* __restrict__ img,
                                                 float* __restrict__ out_acc,
                                                 const float* __restrict__ gAll,
                                                 int gOff, int accumulate) {
    constexpr int PAD   = K / 2;
    constexpr int NCV   = (K + 18) / 4;       // chunks per 16-row subtile
    constexpr int RT    = 16 + 4 * NCV;       // staged ext rows (covers wy=0,1)
    constexpr int VSTR  = 72;                 // 64 cols + bank-split pad
    constexpr int GPADN = 4 * NCV + 18;

    __shared__ float lds[RT * VSTR];
    __shared__ float gpad[GPADN];

    const int tid  = threadIdx.x;
    const int lane = tid & 31;
    const int wave = tid >> 5;
    const int half = (lane >> 4) & 1;
    const int l    = lane & 15;
    const int wy   = wave >> 2;               // 0..1
    const int wx   = wave & 3;                // 0..3

    const int plane = blockIdx.z;
    const int y0    = blockIdx.y * 32;
    const int xbase = blockIdx.x * 64;
    const float* src  = tmp + (size_t)plane * IMG_H * IMG_W;
    const float* srci = img + (size_t)plane * IMG_H * IMG_W;
    float*       dst  = out_acc + (size_t)plane * IMG_H * IMG_W;

    for (int i = tid; i < GPADN; i += 256) {
        const int j = i - 16;
        gpad[i] = (j >= 0 && j < K) ? gAll[gOff + j] : 0.0f;
    }
    // stage RT rows x 64 cols with vertical reflect padding
    for (int i = tid; i < RT * 64; i += 256) {
        const int r  = i >> 6;
        const int cl = i & 63;
        const int gy = reflect_idx(y0 - PAD + r, IMG_H);
        lds[r * VSTR + cl] = src[(size_t)gy * IMG_W + xbase + cl];
    }
    __syncthreads();

    const int rbase = wy * 16;
    const int cbase = wx * 16;
    const int bbase = (rbase + 2 * half) * VSTR + cbase + l;  // + 4*ch*VSTR
    const int wbase = 2 * half - l + 16;                      // + 4*ch (gpad)

    auto loadA = [&](int ch) {                // A[m=l, p] = g[p - l], padded
        v2f a;
        a.x = gpad[wbase + 4 * ch];
        a.y = gpad[wbase + 4 * ch + 1];
        return a;
    };
    auto loadB = [&](int ch) {
        v2f b;
        b.x = lds[bbase + 4 * ch * VSTR];
        b.y = lds[bbase + 4 * ch * VSTR + VSTR];
        return b;
    };

    v8f acc = {};
    v2f a = loadA(0), b = loadB(0);
#pragma unroll 4
    for (int ch = 0; ch < NCV - 1; ++ch) {
        v2f an = loadA(ch + 1);
        v2f bn = loadB(ch + 1);
        acc = __builtin_amdgcn_wmma_f32_16x16x4_f32(false, a, false, b,
                                                    (short)0, acc, false, false);
        a = an; b = bn;
    }
    acc = __builtin_amdgcn_wmma_f32_16x16x4_f32(false, a, false, b,
                                                (short)0, acc, false, false);

    const int gx = xbase + cbase + l;
#pragma unroll
    for (int r = 0; r < 8; ++r) {
        const int gy = y0 + rbase + r + 8 * half;
        const size_t idx = (size_t)gy * IMG_W + gx;
        const float ill = acc[r] + 1e-6f;               // blur(img+1e-6) + 1e-6
        const float val = logf(srci[idx] + 1e-6f) - logf(ill);
        if (accumulate) dst[idx] += val;
        else            dst[idx]  = val;
    }
}

// ---------------------------------------------------------------------------
// Global min/max (two-stage, deterministic, b128 loads) + in-place normalize.
// retinex = acc/3, but the /3 cancels in (r-min)/(max-min), so skip it.
// ---------------------------------------------------------------------------
__global__ __launch_bounds__(256) void msr_minmax_part(const float4* __restrict__ a,
                                                       long long n4,
                                                       float* __restrict__ part) {
    __shared__ float smn[256], smx[256];
    float mn = 3.4e38f, mx = -3.4e38f;
    long long i = (long long)blockIdx.x * 256 + threadIdx.x;
    const long long stride = (long long)gridDim.x * 256;
    for (; i < n4; i += stride) {
        const float4 v = a[i];
        mn = fminf(mn, fminf(fminf(v.x, v.y), fminf(v.z, v.w)));
        mx = fmaxf(mx, fmaxf(fmaxf(v.x, v.y), fmaxf(v.z, v.w)));
    }
    smn[threadIdx.x] = mn; smx[threadIdx.x] = mx;
    __syncthreads();
    for (int st = 128; st > 0; st >>= 1) {
        if ((int)threadIdx.x < st) {
            smn[threadIdx.x] = fminf(smn[threadIdx.x], smn[threadIdx.x + st]);
            smx[threadIdx.x] = fmaxf(smx[threadIdx.x], smx[threadIdx.x + st]);
        }
        __syncthreads();
    }
    if (threadIdx.x == 0) {
        part[blockIdx.x] = smn[0];
        part[gridDim.x + blockIdx.x] = smx[0];
    }
}

__global__ __launch_bounds__(256) void msr_minmax_final(const float* __restrict__ part,
                                                        int nblk,
                                                        float* __restrict__ mm) {
    __shared__ float smn[256], smx[256];
    float mn = 3.4e38f, mx = -3.4e38f;
    for (int i = threadIdx.x; i < nblk; i += 256) {
        mn = fminf(mn, part[i]);
        mx = fmaxf(mx, part[nblk + i]);
    }
    smn[threadIdx.x] = mn; smx[threadIdx.x] = mx;
    __syncthreads();
    for (int st = 128; st > 0; st >>= 1) {
        if ((int)threadIdx.x < st) {
            smn[threadIdx.x] = fminf(smn[threadIdx.x], smn[threadIdx.x + st]);
            smx[threadIdx.x] = fmaxf(smx[threadIdx.x], smx[threadIdx.x + st]);
        }
        __syncthreads();
    }
    if (threadIdx.x == 0) { mm[0] = smn[0]; mm[1] = smx[0]; }
}

__global__ __launch_bounds__(256) void msr_normalize(float4* __restrict__ out,
                                                     const float* __restrict__ mm,
                                                     long long n4) {
    const float mn = mm[0];
    const float sc = 255.0f / (mm[1] - mn);
    long long i = (long long)blockIdx.x * 256 + threadIdx.x;
    const long long stride = (long long)gridDim.x * 256;
    for (; i < n4; i += stride) {
        float4 v = out[i];
        v.x = (v.x - mn) * sc;
        v.y = (v.y - mn) * sc;
        v.z = (v.z - mn) * sc;
        v.w = (v.w - mn) * sc;
        out[i] = v;
    }
}

// ---------------------------------------------------------------------------
// Launch. Workspace layout (floats):
//   [0,512)       gaussian coeffs (273 used)
//   [1024,5120)   min/max partials (2*2048)
//   [5120,5122)   final {min,max}
//   [16384, ...)  horizontal-pass tmp image (N_ELEMS floats, ~100.7 MB)
// ---------------------------------------------------------------------------
extern "C" void kernel_launch(void* const* d_in, const int* in_sizes, int n_in,
                              void* d_out, int out_size, void* d_ws, size_t ws_size,
                              hipStream_t stream) {
    (void)in_sizes; (void)n_in; (void)out_size; (void)ws_size;
    const float* img = (const float*)d_in[0];
    // d_in[1] (sigmas) is {15,40,80} per the reference setup; kernel sizes must
    // be compile-time static (templates), mirroring the reference's constraint.
    float* out = (float*)d_out;

    float* ws    = (float*)d_ws;
    float* gAll  = ws;
    float* part  = ws + 1024;
    float* mm    = ws + 5120;
    float* tmp   = ws + 16384;

    msr_gauss_init<<<1, 256, 0, stream>>>(gAll);

    const dim3 hgrid(IMG_W / 128, IMG_H / 16, N_PLANES);
    const dim3 vgrid(IMG_W / 64,  IMG_H / 32, N_PLANES);

    // scale 0: K=31 (sigma 15), off 0
    msr_hblur<31><<<hgrid, 256, 0, stream>>>(img, tmp, gAll, 0);
    msr_vblur<31><<<vgrid, 256, 0, stream>>>(tmp, img, out, gAll, 0, 0);
    // scale 1: K=81 (sigma 40), off 31
    msr_hblur<81><<<hgrid, 256, 0, stream>>>(img, tmp, gAll, 31);
    msr_vblur<81><<<vgrid, 256, 0, stream>>>(tmp, img, out, gAll, 31, 1);
    // scale 2: K=161 (sigma 80), off 112
    msr_hblur<161><<<hgrid, 256, 0, stream>>>(img, tmp, gAll, 112);
    msr_vblur<161><<<vgrid, 256, 0, stream>>>(tmp, img, out, gAll, 112, 1);

    msr_minmax_part<<<2048, 256, 0, stream>>>((const float4*)out, N_ELEMS / 4, part);
    msr_minmax_final<<<1, 256, 0, stream>>>(part, 2048, mm);
    msr_normalize<<<8192, 256, 0, stream>>>((float4*)out, mm, N_ELEMS / 4);
}